// LuongAttnDecoderRNN_14851996909710
// MI455X (gfx1250) — compile-verified
//
#include <hip/hip_runtime.h>

typedef __attribute__((ext_vector_type(2))) float v2f;
typedef __attribute__((ext_vector_type(8))) float v8f;

#define B_ 64
#define S_ 128
#define H_ 1024
#define V_ 32000

// ---------------------------------------------------------------------------
// Generic fp32 WMMA GEMM: D[M,N] = act(A[M,K] * op(B) + bias)
//   TRANS_B=1 : B given as [N,K] row-major (NT)  -> contiguous float2 loads
//   TRANS_B=0 : B given as [K,N] row-major (NN)  -> strided loads
//   ACT=1     : tanh epilogue
//   GATHER_A  : A row b comes from A[gather[b], :] (embedding lookup)
// One wave computes one 16x16 tile via V_WMMA_F32_16X16X4_F32 (K step 4).
// ISA 7.12.2 layouts: A/B lane holds K = 2*(lane>>4)+v at M/N = lane&15;
// C vgpr r holds row M = r + 8*(lane>>4).
// ---------------------------------------------------------------------------
template <bool TRANS_B, int ACT, bool GATHER_A>
__global__ __launch_bounds__(256) void wmma_gemm_f32(
    const float* __restrict__ A, int lda, const float* __restrict__ Bm, int ldb,
    const float* __restrict__ bias, float* __restrict__ D, int ldd, int M,
    int N, int K, const int* __restrict__ gather) {
  const int lane = threadIdx.x & 31;
  const int wid = (blockIdx.x * blockDim.x + threadIdx.x) >> 5;  // wave-uniform
  const int Mt = M >> 4, Nt = N >> 4;
  if (wid >= Mt * Nt) return;  // whole-wave exit -> EXEC all ones below
  const int row0 = (wid % Mt) << 4;
  const int col0 = (wid / Mt) << 4;
  const int l15 = lane & 15;
  const int hi = (lane >> 4) << 1;  // 0 or 2
  const int mrow = row0 + l15;
  const int ncol = col0 + l15;
  const float* arow = GATHER_A ? (A + (size_t)gather[mrow] * lda)
                               : (A + (size_t)mrow * lda);
  const float* bptr = TRANS_B ? (Bm + (size_t)ncol * ldb) : (Bm + ncol);

  v8f c = {};
  for (int k = 0; k < K; k += 4) {
    v2f a = *(const v2f*)(arow + k + hi);
    v2f b;
    if (TRANS_B) {
      b = *(const v2f*)(bptr + k + hi);
    } else {
      b.x = bptr[(size_t)(k + hi) * ldb];
      b.y = bptr[(size_t)(k + hi + 1) * ldb];
    }
    c = __builtin_amdgcn_wmma_f32_16x16x4_f32(false, a, false, b, (short)0, c,
                                              false, false);
  }

  const float bb = bias ? bias[ncol] : 0.0f;
  const int mbase = row0 + ((lane >> 4) << 3);
#pragma unroll
  for (int r = 0; r < 8; ++r) {
    float v = c[r] + bb;
    if (ACT == 1) v = tanhf(v);
    D[(size_t)(mbase + r) * ldd + ncol] = v;
  }
}

// ---------------------------------------------------------------------------
// GRU gates (PyTorch ordering r,z,n): h_new = (1-z)*n + z*h
// Writes h_new into concat buffer (stride 2H, left half) and into d_out.
// ---------------------------------------------------------------------------
__global__ __launch_bounds__(256) void gru_gate_kernel(
    const float* __restrict__ gi, const float* __restrict__ gh,
    const float* __restrict__ h, float* __restrict__ cat,
    float* __restrict__ h_out) {
  const int idx = blockIdx.x * blockDim.x + threadIdx.x;
  if (idx >= B_ * H_) return;
  const int b = idx >> 10;        // / H_
  const int j = idx & (H_ - 1);   // % H_
  const size_t base = (size_t)b * 3 * H_;
  const float sr = gi[base + j] + gh[base + j];
  const float sz = gi[base + H_ + j] + gh[base + H_ + j];
  const float r = 1.0f / (1.0f + expf(-sr));
  const float z = 1.0f / (1.0f + expf(-sz));
  const float n = tanhf(gi[base + 2 * H_ + j] + r * gh[base + 2 * H_ + j]);
  const float hv = h[(size_t)b * H_ + j];
  const float hn = (1.0f - z) * n + z * hv;
  cat[(size_t)b * 2 * H_ + j] = hn;
  h_out[(size_t)b * H_ + j] = hn;
}

// ---------------------------------------------------------------------------
// energies[b,s] = hp[b,:] . enc[s,b,:]   (one wave per (b,s), float4 loads)
// (attn_b contributes a per-b constant -> cancelled by softmax; omitted)
// ---------------------------------------------------------------------------
__global__ __launch_bounds__(256) void energies_kernel(
    const float* __restrict__ hp, const float* __restrict__ enc,
    float* __restrict__ energ) {
  const int lane = threadIdx.x & 31;
  const int wid = (blockIdx.x * blockDim.x + threadIdx.x) >> 5;
  if (wid >= B_ * S_) return;
  const int b = wid >> 7;         // / S_
  const int s = wid & (S_ - 1);
  const float4* hpb = (const float4*)(hp + (size_t)b * H_);
  const float4* eb = (const float4*)(enc + ((size_t)s * B_ + b) * H_);
  float acc = 0.0f;
  for (int i = lane; i < H_ / 4; i += 32) {
    const float4 x = hpb[i], y = eb[i];
    acc += x.x * y.x + x.y * y.y + x.z * y.z + x.w * y.w;
  }
#pragma unroll
  for (int off = 16; off; off >>= 1) acc += __shfl_down(acc, off, 32);
  if (lane == 0) energ[b * S_ + s] = acc;
}

// ---------------------------------------------------------------------------
// softmax over S per batch row; writes weights in-place and to d_out slice
// ---------------------------------------------------------------------------
__global__ __launch_bounds__(S_) void softmax_kernel(
    float* __restrict__ energ, float* __restrict__ attn_out) {
  const int b = blockIdx.x;
  const int s = threadIdx.x;
  __shared__ float sm[S_];
  const float e = energ[b * S_ + s];
  sm[s] = e;
  __syncthreads();
  for (int off = S_ / 2; off; off >>= 1) {
    if (s < off) sm[s] = fmaxf(sm[s], sm[s + off]);
    __syncthreads();
  }
  const float mx = sm[0];
  __syncthreads();
  const float ex = expf(e - mx);
  sm[s] = ex;
  __syncthreads();
  for (int off = S_ / 2; off; off >>= 1) {
    if (s < off) sm[s] += sm[s + off];
    __syncthreads();
  }
  const float w = ex / sm[0];
  energ[b * S_ + s] = w;
  attn_out[b * S_ + s] = w;
}

// ---------------------------------------------------------------------------
// context[b,h] = sum_s w[b,s] * enc[s,b,h]; written to right half of concat
// buffer (offset H, row stride 2H). 256 threads x float4 covers H=1024.
// ---------------------------------------------------------------------------
__global__ __launch_bounds__(256) void context_kernel(
    const float* __restrict__ attn, const float* __restrict__ enc,
    float* __restrict__ cat) {
  const int b = blockIdx.x;
  __shared__ float w[S_];
  for (int s = threadIdx.x; s < S_; s += blockDim.x) w[s] = attn[b * S_ + s];
  __syncthreads();
  const int h0 = threadIdx.x * 4;  // 256*4 == H_
  float4 acc = {0.f, 0.f, 0.f, 0.f};
  for (int s = 0; s < S_; ++s) {
    const float4 e = *(const float4*)(enc + ((size_t)s * B_ + b) * H_ + h0);
    const float ws = w[s];
    acc.x += ws * e.x;
    acc.y += ws * e.y;
    acc.z += ws * e.z;
    acc.w += ws * e.w;
  }
  *(float4*)(cat + (size_t)b * 2 * H_ + H_ + h0) = acc;
}

// ---------------------------------------------------------------------------
extern "C" void kernel_launch(void* const* d_in, const int* in_sizes, int n_in,
                              void* d_out, int out_size, void* d_ws,
                              size_t ws_size, hipStream_t stream) {
  (void)in_sizes; (void)n_in; (void)out_size; (void)ws_size;
  const int* input_seq = (const int*)d_in[0];
  const float* last_hidden = (const float*)d_in[1];   // [1,B,H] -> [B,H]
  const float* enc = (const float*)d_in[2];           // [S,B,H]
  const float* emb = (const float*)d_in[3];           // [V,H]
  const float* W_ih = (const float*)d_in[4];          // [3H,H]
  const float* W_hh = (const float*)d_in[5];          // [3H,H]
  const float* b_ih = (const float*)d_in[6];
  const float* b_hh = (const float*)d_in[7];
  const float* attn_W = (const float*)d_in[8];        // [H,H]
  // d_in[9] = attn_b: per-b constant in energies, cancelled by softmax
  const float* concat_W = (const float*)d_in[10];     // [H,2H]
  const float* concat_b = (const float*)d_in[11];
  const float* out_W = (const float*)d_in[12];        // [V,H]
  const float* out_b = (const float*)d_in[13];

  float* out = (float*)d_out;
  float* out_main = out;                              // [B,V]
  float* out_h = out + (size_t)B_ * V_;               // [1,B,H]
  float* out_attn = out_h + (size_t)B_ * H_;          // [B,1,S]

  float* ws = (float*)d_ws;
  float* gi = ws;                         // B*3H
  float* gh = gi + (size_t)B_ * 3 * H_;   // B*3H
  float* cat = gh + (size_t)B_ * 3 * H_;  // B*2H  (h_new | context)
  float* hp = cat + (size_t)B_ * 2 * H_;  // B*H   (h_new @ attn_W)
  float* energ = hp + (size_t)B_ * H_;    // B*S
  float* cout = energ + (size_t)B_ * S_;  // B*H

  auto grid_for = [](int tiles) { return (tiles * 32 + 255) / 256; };

  // 1) gi = emb[input_seq] @ W_ih^T + b_ih   (A gathered, B NT)
  wmma_gemm_f32<true, 0, true><<<grid_for((B_ / 16) * (3 * H_ / 16)), 256, 0,
                                 stream>>>(emb, H_, W_ih, H_, b_ih, gi, 3 * H_,
                                           B_, 3 * H_, H_, input_seq);
  // 2) gh = h @ W_hh^T + b_hh
  wmma_gemm_f32<true, 0, false><<<grid_for((B_ / 16) * (3 * H_ / 16)), 256, 0,
                                  stream>>>(last_hidden, H_, W_hh, H_, b_hh, gh,
                                            3 * H_, B_, 3 * H_, H_, nullptr);
  // 3) GRU gates -> h_new (into concat buffer left half + d_out)
  gru_gate_kernel<<<(B_ * H_) / 256, 256, 0, stream>>>(gi, gh, last_hidden,
                                                       cat, out_h);
  // 4) hp = h_new @ attn_W   (A = concat buffer, lda=2H; B NN)
  wmma_gemm_f32<false, 0, false><<<grid_for((B_ / 16) * (H_ / 16)), 256, 0,
                                   stream>>>(cat, 2 * H_, attn_W, H_, nullptr,
                                             hp, H_, B_, H_, H_, nullptr);
  // 5) energies[b,s] = hp[b] . enc[s,b]
  energies_kernel<<<grid_for(B_ * S_), 256, 0, stream>>>(hp, enc, energ);
  // 6) softmax over S (in-place + d_out slice)
  softmax_kernel<<<B_, S_, 0, stream>>>(energ, out_attn);
  // 7) context -> concat buffer right half
  context_kernel<<<B_, 256, 0, stream>>>(energ, enc, cat);
  // 8) concat_out = tanh(cat @ concat_W^T + concat_b)
  wmma_gemm_f32<true, 1, false><<<grid_for((B_ / 16) * (H_ / 16)), 256, 0,
                                  stream>>>(cat, 2 * H_, concat_W, 2 * H_,
                                            concat_b, cout, H_, B_, H_,
                                            2 * H_, nullptr);
  // 9) output = concat_out @ out_W^T + out_b   (the 131 MB HBM-bound GEMM)
  wmma_gemm_f32<true, 0, false><<<grid_for((B_ / 16) * (V_ / 16)), 256, 0,
                                  stream>>>(cout, H_, out_W, H_, out_b,
                                            out_main, V_, B_, V_, H_, nullptr);
}